// PixelVectorExtractor_86577950752801
// MI455X (gfx1250) — compile-verified
//
#include <hip/hip_runtime.h>
#include <hip/hip_bf16.h>

typedef __attribute__((ext_vector_type(2))) float v2f;
typedef __attribute__((ext_vector_type(8))) float v8f;

// ---- f32 16x16x4 WMMA fragment helpers (wave32) ----------------------------
// A (16x4, MxK): lane<16 -> row=lane, K={k0,k0+1}; lane>=16 -> row=lane-16, K={k0+2,k0+3}
__device__ __forceinline__ v2f lda_f(const float* p, int ld, int r0, int k0, int lane) {
    int m = lane & 15, kb = (lane >> 4) << 1;
    const float* q = p + (r0 + m) * ld + k0 + kb;
    v2f r; r.x = q[0]; r.y = q[1];
    return r;
}
// B (4x16, KxN): lane<16 -> col=lane, K={k0,k0+1}; lane>=16 -> col=lane-16, K={k0+2,k0+3}
__device__ __forceinline__ v2f ldb_f(const float* p, int ld, int k0, int n0, int lane) {
    int n = lane & 15, kb = (lane >> 4) << 1;
    v2f r;
    r.x = p[(k0 + kb)     * ld + n0 + n];
    r.y = p[(k0 + kb + 1) * ld + n0 + n];
    return r;
}
// C/D (16x16 f32): VGPR g holds row g (lanes 0-15) / row g+8 (lanes 16-31), col = lane&15
__device__ __forceinline__ v8f ldc_f(const float* p, int ld, int r0, int c0, int lane) {
    int col = lane & 15, rb = (lane >> 4) << 3;
    v8f c;
#pragma unroll
    for (int g = 0; g < 8; ++g) c[g] = p[(r0 + rb + g) * ld + c0 + col];
    return c;
}
__device__ __forceinline__ void stc_f(float* p, int ld, int r0, int c0, int lane, v8f c) {
    int col = lane & 15, rb = (lane >> 4) << 3;
#pragma unroll
    for (int g = 0; g < 8; ++g) p[(r0 + rb + g) * ld + c0 + col] = c[g];
}
__device__ __forceinline__ v8f wmma4(v2f a, v2f b, v8f c) {
    return __builtin_amdgcn_wmma_f32_16x16x4_f32(false, a, false, b, (short)0, c, false, false);
}

// LayerNorm over the 11 real dims of a 16-wide row; zero the padding dims.
__device__ __forceinline__ void ln_row(float* row, const float* w) {
    float m = 0.f;
#pragma unroll
    for (int c = 0; c < 11; ++c) m += row[c];
    m *= (1.0f / 11.0f);
    float v = 0.f;
#pragma unroll
    for (int c = 0; c < 11; ++c) { float d = row[c] - m; v += d * d; }
    v *= (1.0f / 11.0f);
    float rinv = rsqrtf(v + 1e-5f);
#pragma unroll
    for (int c = 0; c < 16; ++c) row[c] = (c < 11) ? (row[c] - m) * rinv * w[c] : 0.f;
}

// Problem constants (reference setup): N=1, C=10, H=W=16, D=11, S=900, FF=64.
// Per sequence we keep 64 tokens: 49 window tokens + zero tokens (token 49 is the
// representative for the 851 all-zero canvas positions).
__global__ __launch_bounds__(128)
void pixel_encoder_kernel(const float* __restrict__ x,     // [10,16,16]
                          const float* __restrict__ Wqkv,  // [33,11]
                          const float* __restrict__ Wo,    // [11,11]
                          const float* __restrict__ W1,    // [64,11]
                          const float* __restrict__ W2,    // [11,64]
                          const float* __restrict__ ln1w,  // [11]
                          const float* __restrict__ ln2w,  // [11]
                          float* __restrict__ out)         // [256,10,900]
{
    __shared__ float Wq_t[16 * 16], Wk_t[16 * 16], Wv_t[16 * 16], Wo_t[16 * 16];
    __shared__ float W1_t[16 * 64];   // B[k][n] = W1[n][k], k<16(pad), n<64
    __shared__ float W2_t[64 * 16];   // B[k][n] = W2[n][k], k<64, n<16(pad)
    __shared__ float ln1[16], ln2[16];
    __shared__ float Xb[64 * 16];     // token matrix (and later residual1 / X1)
    __shared__ float Qb[64 * 16], Kb[64 * 16], Vb[64 * 16];
    __shared__ float Pb[64 * 64];     // scores/probs, reused for FF hidden
    __shared__ float Ab[64 * 16];     // attn out, reused for residual2 / final

    const int tid  = threadIdx.x;
    const int lane = tid & 31;
    const int wv   = tid >> 5;
    const int r0   = wv * 16;
    const int p    = blockIdx.x;      // pixel / sequence index
    const int hh   = p >> 4;
    const int ww   = p & 15;

    // ---- stage padded weights (scale folded into Wq) ----
    const float scale = rsqrtf(11.0f);
    for (int i = tid; i < 256; i += 128) {
        int k = i >> 4, n = i & 15;
        float q = 0.f, kk = 0.f, vv = 0.f, o = 0.f;
        if (n < 11 && k < 11) {
            q  = Wqkv[n * 11 + k] * scale;
            kk = Wqkv[(11 + n) * 11 + k];
            vv = Wqkv[(22 + n) * 11 + k];
            o  = Wo[n * 11 + k];
        }
        Wq_t[i] = q; Wk_t[i] = kk; Wv_t[i] = vv; Wo_t[i] = o;
    }
    for (int i = tid; i < 1024; i += 128) {
        int k1 = i >> 6, n1 = i & 63;               // W1_t[k][n]
        W1_t[i] = (k1 < 11) ? W1[n1 * 11 + k1] : 0.f;
        int k2 = i >> 4, n2 = i & 15;               // W2_t[k][n]
        W2_t[i] = (n2 < 11) ? W2[n2 * 64 + k2] : 0.f;
    }
    if (tid < 16) {
        ln1[tid] = (tid < 11) ? ln1w[tid] : 0.f;
        ln2[tid] = (tid < 11) ? ln2w[tid] : 0.f;
    }

    // ---- build token matrix X [64 x 16] ----
    for (int i = tid; i < 1024; i += 128) {
        int t = i >> 4, ch = i & 15;
        float val = 0.f;
        if (t < 49) {
            int wi = t / 7, wj = t % 7;
            int gy = hh + wi - 3, gx = ww + wj - 3;
            bool inb = ((unsigned)gy < 16u) && ((unsigned)gx < 16u);
            if (ch < 10)       val = inb ? x[ch * 256 + gy * 16 + gx] : 0.f;
            else if (ch == 10) val = inb ? 0.f : 1.f;   // pad-mask channel
        }
        Xb[i] = val;
    }
    __syncthreads();

    // ---- QKV: [64,16] x [16,16] per output ----
    {
        v8f cq = {}, ck = {}, cv = {};
#pragma unroll
        for (int kk = 0; kk < 16; kk += 4) {
            v2f a = lda_f(Xb, 16, r0, kk, lane);
            cq = wmma4(a, ldb_f(Wq_t, 16, kk, 0, lane), cq);
            ck = wmma4(a, ldb_f(Wk_t, 16, kk, 0, lane), ck);
            cv = wmma4(a, ldb_f(Wv_t, 16, kk, 0, lane), cv);
        }
        stc_f(Qb, 16, r0, 0, lane, cq);
        stc_f(Kb, 16, r0, 0, lane, ck);
        stc_f(Vb, 16, r0, 0, lane, cv);
    }
    __syncthreads();

    // ---- S = Q * K^T  (K^T fragment == A-style load from row-major K) ----
#pragma unroll
    for (int ct = 0; ct < 4; ++ct) {
        v8f c = {};
#pragma unroll
        for (int kk = 0; kk < 16; kk += 4) {
            v2f a = lda_f(Qb, 16, r0, kk, lane);
            v2f b = lda_f(Kb, 16, ct * 16, kk, lane);
            c = wmma4(a, b, c);
        }
        stc_f(Pb, 64, r0, ct * 16, lane, c);
    }
    __syncthreads();

    // ---- softmax rows, with 836 out-of-tile zero keys folded in ----
    if (tid < 64) {
        float* row = Pb + tid * 64;
        float m = 0.f;                         // zero keys contribute logit 0
        for (int j = 0; j < 64; ++j) m = fmaxf(m, row[j]);
        float s = 836.f * expf(-m);            // 900 keys total, 64 in-tile
        for (int j = 0; j < 64; ++j) { float e = expf(row[j] - m); row[j] = e; s += e; }
        float inv = 1.f / s;
        for (int j = 0; j < 64; ++j) row[j] *= inv;
    }
    __syncthreads();

    // ---- A = P * V ----
    {
        v8f c = {};
#pragma unroll
        for (int kk = 0; kk < 64; kk += 4)
            c = wmma4(lda_f(Pb, 64, r0, kk, lane), ldb_f(Vb, 16, kk, 0, lane), c);
        stc_f(Ab, 16, r0, 0, lane, c);
    }
    __syncthreads();

    // ---- residual1 = X + A * Wo^T  (C preloaded with X) ----
    {
        v8f c = ldc_f(Xb, 16, r0, 0, lane);
#pragma unroll
        for (int kk = 0; kk < 16; kk += 4)
            c = wmma4(lda_f(Ab, 16, r0, kk, lane), ldb_f(Wo_t, 16, kk, 0, lane), c);
        stc_f(Xb, 16, r0, 0, lane, c);
    }
    __syncthreads();

    // ---- LayerNorm1 (in place on Xb) ----
    if (tid < 64) ln_row(Xb + tid * 16, ln1);
    __syncthreads();

    // ---- H = relu(X1 * W1^T)  [64,64] ----
#pragma unroll
    for (int ct = 0; ct < 4; ++ct) {
        v8f c = {};
#pragma unroll
        for (int kk = 0; kk < 16; kk += 4)
            c = wmma4(lda_f(Xb, 16, r0, kk, lane), ldb_f(W1_t, 64, kk, ct * 16, lane), c);
#pragma unroll
        for (int g = 0; g < 8; ++g) c[g] = fmaxf(c[g], 0.f);
        stc_f(Pb, 64, r0, ct * 16, lane, c);
    }
    __syncthreads();

    // ---- residual2 = X1 + H * W2^T ----
    {
        v8f c = ldc_f(Xb, 16, r0, 0, lane);
#pragma unroll
        for (int kk = 0; kk < 64; kk += 4)
            c = wmma4(lda_f(Pb, 64, r0, kk, lane), ldb_f(W2_t, 16, kk, 0, lane), c);
        stc_f(Ab, 16, r0, 0, lane, c);
    }
    __syncthreads();

    // ---- LayerNorm2 (in place on Ab) ----
    if (tid < 64) ln_row(Ab + tid * 16, ln2);
    __syncthreads();

    // ---- scatter to [10, 900]: window positions pick their token, the other
    //      851 canvas positions all equal the zero-token (row 49) output ----
    float* ob = out + p * 9000;
    for (int s = tid; s < 900; s += 128) {
        int r = s / 30, cc = s % 30;
        int t = (r < 7 && cc < 7) ? (r * 7 + cc) : 49;
        const float* row = Ab + t * 16;
#pragma unroll
        for (int ch = 0; ch < 10; ++ch) ob[ch * 900 + s] = row[ch];
    }
}

extern "C" void kernel_launch(void* const* d_in, const int* in_sizes, int n_in,
                              void* d_out, int out_size, void* d_ws, size_t ws_size,
                              hipStream_t stream) {
    const float* x    = (const float*)d_in[0];
    const float* Wqkv = (const float*)d_in[1];
    const float* Wo   = (const float*)d_in[2];
    const float* W1   = (const float*)d_in[3];
    const float* W2   = (const float*)d_in[4];
    const float* l1   = (const float*)d_in[5];
    const float* l2   = (const float*)d_in[6];
    (void)in_sizes; (void)n_in; (void)out_size; (void)d_ws; (void)ws_size;
    hipLaunchKernelGGL(pixel_encoder_kernel, dim3(256), dim3(128), 0, stream,
                       x, Wqkv, Wo, W1, W2, l1, l2, (float*)d_out);
}